// TnoFD_12206297055424
// MI455X (gfx1250) — compile-verified
//
#include <hip/hip_runtime.h>
#include <math.h>

// Problem dims (from reference)
#define B_   8
#define H_   8
#define N_   4096
#define D_   64
#define NF_  4097          // rfft bins of 8192
#define FFT_ 8192
#define P_   32            // channel PAIRS per (b,h): pack (2p, 2p+1) as complex
#define LDA_ 130           // padded LDS row stride in floats (bank-conflict free)
#define PI_  3.14159265358979323846f

typedef float v2f __attribute__((ext_vector_type(2)));
typedef float v8f __attribute__((ext_vector_type(8)));

// CDNA5 full-precision matrix core: D(16x16,f32) += A(16x4,f32) x B(4x16,f32)
static __device__ __forceinline__ v8f wmma4(v2f a, v2f b, v8f c) {
  return __builtin_amdgcn_wmma_f32_16x16x4_f32(false, a, false, b, (short)0, c,
                                               false, false);
}
static __device__ __forceinline__ v8f zero8() {
  v8f z = {0.f,0.f,0.f,0.f,0.f,0.f,0.f,0.f};
  return z;
}

// ---------------------------------------------------------------------------
// Kernel 1: RPE filter MLP -> complex filter planes Fre/Fim [H][NF][D]
// One wave32 per 16 frequency rows; activations staged in LDS; all matmuls
// on v_wmma_f32_16x16x4_f32.
// ---------------------------------------------------------------------------
static __device__ void rms_relu16(float (*buf)[68]) {
  const int lane = threadIdx.x;
  if (lane < 16) {
    float s = 0.f;
    for (int j = 0; j < 64; ++j) { float v = buf[lane][j]; s += v * v; }
    float r = rsqrtf(s * (1.0f/64.0f) + 1e-8f);
    for (int j = 0; j < 64; ++j) {
      float v = buf[lane][j] * r;
      buf[lane][j] = v > 0.f ? v : 0.f;
    }
  }
  __syncthreads();
}

static __device__ void mm64x64(const float* __restrict__ W,
                               const float* __restrict__ bias,
                               float (*src)[68], float (*dst)[68]) {
  const int lane = threadIdx.x;
  const int la = lane & 15, hi = lane >> 4;
  #pragma unroll
  for (int nt = 0; nt < 4; ++nt) {
    v8f acc = zero8();
    const int n = nt * 16 + la;
    for (int kc = 0; kc < 16; ++kc) {
      const int r0 = 4 * kc + 2 * hi;
      v2f a = { src[la][r0], src[la][r0 + 1] };          // A: 16x4 fragment
      v2f b = { W[r0 * 64 + n], W[(r0 + 1) * 64 + n] };  // B: 4x16 fragment
      acc = wmma4(a, b, acc);
    }
    #pragma unroll
    for (int e = 0; e < 8; ++e) dst[e + 8 * hi][n] = acc[e] + bias[n];
  }
  __syncthreads();
}

__global__ __launch_bounds__(32) void rpe_kernel(
    const float* __restrict__ w0, const float* __restrict__ b0,
    const float* __restrict__ w1, const float* __restrict__ b1,
    const float* __restrict__ w2, const float* __restrict__ b2,
    const float* __restrict__ w3, const float* __restrict__ b3,
    const float* __restrict__ wo, const float* __restrict__ bo,
    float* __restrict__ Fre, float* __restrict__ Fim)
{
  __shared__ float xb[16][68];
  __shared__ float yb[16][68];
  const int lane = threadIdx.x;
  const int la = lane & 15, hi = lane >> 4;
  const int f0 = blockIdx.x * 16;

  // x0 = w_axis @ w0 + b0   (w_axis[f] = pi*f/4096)
  for (int i = 0; i < 32; ++i) {
    int e = lane + i * 32;             // 0..1023
    int m = e >> 6, j = e & 63;
    float wax = PI_ * (float)(f0 + m) * (1.0f / 4096.0f);
    xb[m][j] = wax * w0[j] + b0[j];
  }
  __syncthreads();

  rms_relu16(xb); mm64x64(w1, b1, xb, yb);
  rms_relu16(yb); mm64x64(w2, b2, yb, xb);
  rms_relu16(xb); mm64x64(w3, b3, xb, yb);
  rms_relu16(yb);

  // final: (16x64) @ wo(64x1024) + bo -> filter planes
  for (int nt = 0; nt < 64; ++nt) {
    v8f acc = zero8();
    const int n = nt * 16 + la;
    for (int kc = 0; kc < 16; ++kc) {
      const int r0 = 4 * kc + 2 * hi;
      v2f a = { yb[la][r0], yb[la][r0 + 1] };
      v2f b = { wo[r0 * 1024 + n], wo[(r0 + 1) * 1024 + n] };
      acc = wmma4(a, b, acc);
    }
    const int h = n >> 7, c = n & 127;
    #pragma unroll
    for (int e = 0; e < 8; ++e) {
      const int f = f0 + e + 8 * hi;
      if (f <= 4096) {
        float v = acc[e] + bo[n];
        if (c < 64) Fre[((size_t)h * NF_ + f) * 64 + c] = v;
        else        Fim[((size_t)h * NF_ + f) * 64 + (c - 64)] =
                        (f == 0 || f == 4096) ? 0.f : v;   // imag forced 0 at ends
      }
    }
  }
}

// ---------------------------------------------------------------------------
// Four-step FFT pieces (64 x 128 complex matrix per workgroup, in LDS planes).
// Right-multiply by DFT128: D = S @ E128(sgn), E[r][c] = exp(sgn*2pi*i*r*c/128)
// ---------------------------------------------------------------------------
static __device__ void dft_right(const float* __restrict__ Sre,
                                 const float* __restrict__ Sim,
                                 float* __restrict__ Dre,
                                 float* __restrict__ Dim, float sgn)
{
  const int lane = threadIdx.x & 31, wv = threadIdx.x >> 5;
  const int la = lane & 15, hi2 = (lane >> 4) * 2;
  const int col = wv * 16 + la;
  v8f cr[4], ci[4];
  #pragma unroll
  for (int mt = 0; mt < 4; ++mt) { cr[mt] = zero8(); ci[mt] = zero8(); }
  for (int kc = 0; kc < 32; ++kc) {
    const int r0 = 4 * kc + hi2, r1 = r0 + 1;
    float s0, c0, s1, c1;
    __sincosf((2.0f * PI_ / 128.0f) * (float)((r0 * col) & 127), &s0, &c0);
    __sincosf((2.0f * PI_ / 128.0f) * (float)((r1 * col) & 127), &s1, &c1);
    v2f br  = { c0, c1 };
    v2f bi  = { sgn * s0, sgn * s1 };
    v2f bin = { -bi.x, -bi.y };
    #pragma unroll
    for (int mt = 0; mt < 4; ++mt) {
      const int m = mt * 16 + la;
      v2f ar = { Sre[m * LDA_ + r0], Sre[m * LDA_ + r1] };
      v2f ai = { Sim[m * LDA_ + r0], Sim[m * LDA_ + r1] };
      cr[mt] = wmma4(ar, br,  cr[mt]);   // +Ar*Br
      cr[mt] = wmma4(ai, bin, cr[mt]);   // -Ai*Bi
      ci[mt] = wmma4(ar, bi,  ci[mt]);   // +Ar*Bi
      ci[mt] = wmma4(ai, br,  ci[mt]);   // +Ai*Br
    }
  }
  #pragma unroll
  for (int mt = 0; mt < 4; ++mt) {
    #pragma unroll
    for (int e = 0; e < 8; ++e) {
      const int m = mt * 16 + e + ((lane < 16) ? 0 : 8);
      Dre[m * LDA_ + col] = cr[mt][e];
      Dim[m * LDA_ + col] = ci[mt][e];
    }
  }
}

// Left-multiply by DFT64: D = A64(sgn) @ S, A[m][k] = exp(sgn*2pi*i*m*k/64).
// KC = K/4 chunks (forward: zero-padded input -> KC=8); MT = output row tiles
// (inverse: only first 32 output rows needed -> MT=2).
template<int KC, int MT>
static __device__ void dft_left(const float* __restrict__ Sre,
                                const float* __restrict__ Sim,
                                float* __restrict__ Dre,
                                float* __restrict__ Dim, float sgn)
{
  const int lane = threadIdx.x & 31, wv = threadIdx.x >> 5;
  const int la = lane & 15, hi2 = (lane >> 4) * 2;
  const int col = wv * 16 + la;
  v8f cr[MT], ci[MT];
  #pragma unroll
  for (int mt = 0; mt < MT; ++mt) { cr[mt] = zero8(); ci[mt] = zero8(); }
  for (int kc = 0; kc < KC; ++kc) {
    const int r0 = 4 * kc + hi2, r1 = r0 + 1;
    v2f br = { Sre[r0 * LDA_ + col], Sre[r1 * LDA_ + col] };
    v2f bi = { Sim[r0 * LDA_ + col], Sim[r1 * LDA_ + col] };
    #pragma unroll
    for (int mt = 0; mt < MT; ++mt) {
      const int a = mt * 16 + la;
      float s0, c0, s1, c1;
      __sincosf((2.0f * PI_ / 64.0f) * (float)((a * r0) & 63), &s0, &c0);
      __sincosf((2.0f * PI_ / 64.0f) * (float)((a * r1) & 63), &s1, &c1);
      v2f ar  = { c0, c1 };
      v2f ai  = { sgn * s0, sgn * s1 };
      v2f ain = { -ai.x, -ai.y };
      cr[mt] = wmma4(ar,  br, cr[mt]);
      cr[mt] = wmma4(ain, bi, cr[mt]);
      ci[mt] = wmma4(ar,  bi, ci[mt]);
      ci[mt] = wmma4(ai,  br, ci[mt]);
    }
  }
  #pragma unroll
  for (int mt = 0; mt < MT; ++mt) {
    #pragma unroll
    for (int e = 0; e < 8; ++e) {
      const int m = mt * 16 + e + ((lane < 16) ? 0 : 8);
      Dre[m * LDA_ + col] = cr[mt][e];
      Dim[m * LDA_ + col] = ci[mt][e];
    }
  }
}

// Elementwise 8192-point twiddle on [row k1][col j]: *= exp(sgn*2pi*i*k1*j/8192)
static __device__ void twiddle8192(float* __restrict__ Re,
                                   float* __restrict__ Im, float sgn)
{
  const int tid = threadIdx.x;
  for (int i = 0; i < 32; ++i) {
    const int e = tid + i * 256;
    const int k1 = e >> 7, j = e & 127;
    const int ph = (k1 * j) & (FFT_ - 1);
    float s, c;
    __sincosf((2.0f * PI_ / (float)FFT_) * (float)ph, &s, &c);
    s *= sgn;
    const float re = Re[k1 * LDA_ + j], im = Im[k1 * LDA_ + j];
    Re[k1 * LDA_ + j] = re * c - im * s;
    Im[k1 * LDA_ + j] = re * s + im * c;
  }
}

// ---------------------------------------------------------------------------
// Kernel 2/4: complex FFT length 8192 via four-step (64 x 128), one workgroup
// per (b,h,channel-pair). INV=0: x -> Z (zero-padded, K=32 in DFT64).
// INV=1: Z -> out (only first 4096 samples, M=32 in DFT64), scaled 1/8192.
// ---------------------------------------------------------------------------
template<int INV>
__global__ __launch_bounds__(256) void fft_kernel(float2* __restrict__ Z,
                                                  const float* __restrict__ x,
                                                  float* __restrict__ out)
{
  extern __shared__ float smem[];
  float* Are = smem;
  float* Aim = Are + 64 * LDA_;
  float* Bre = Aim + 64 * LDA_;
  float* Bim = Bre + 64 * LDA_;
  const int tid = threadIdx.x;
  const int pi = blockIdx.x;          // (b*H + h)*P_ + p
  const int bh = pi >> 5;
  const int p  = pi & (P_ - 1);
  const size_t zb = (size_t)pi * FFT_;

  if (INV == 0) {
    // load packed complex z[n] = x[..,2p] + i x[..,2p+1]; n = n1*128+n2 < 4096
    const float* xp = x + (size_t)bh * N_ * D_;
    for (int i = 0; i < 16; ++i) {
      const int e = tid + i * 256;                 // < 4096
      const int n1 = e >> 7, n2 = e & 127;
      const float2 v = *(const float2*)(xp + (size_t)e * D_ + 2 * p);
      Are[n1 * LDA_ + n2] = v.x;
      Aim[n1 * LDA_ + n2] = v.y;
    }
    __syncthreads();
    dft_left<8, 4>(Are, Aim, Bre, Bim, -1.0f);     // DFT64 over n1 (K=32)
    __syncthreads();
    twiddle8192(Bre, Bim, -1.0f);
    __syncthreads();
    dft_right(Bre, Bim, Are, Aim, -1.0f);          // DFT128 over n2
    __syncthreads();
    // Z[k2*64 + k1] = M[k1][k2]
    for (int i = 0; i < 32; ++i) {
      const int e = tid + i * 256;
      const int k1 = e & 63, k2 = e >> 6;
      Z[zb + e] = make_float2(Are[k1 * LDA_ + k2], Aim[k1 * LDA_ + k2]);
    }
  } else {
    for (int i = 0; i < 32; ++i) {
      const int e = tid + i * 256;
      const int k1 = e & 63, k2 = e >> 6;
      const float2 v = Z[zb + e];
      Are[k1 * LDA_ + k2] = v.x;
      Aim[k1 * LDA_ + k2] = v.y;
    }
    __syncthreads();
    dft_right(Are, Aim, Bre, Bim, 1.0f);           // IDFT128 over k2
    __syncthreads();
    twiddle8192(Bre, Bim, 1.0f);
    __syncthreads();
    dft_left<16, 2>(Bre, Bim, Are, Aim, 1.0f);     // IDFT64, rows a<32 only
    __syncthreads();
    float* op = out + (size_t)bh * N_ * D_;
    const float sc = 1.0f / (float)FFT_;
    for (int i = 0; i < 16; ++i) {
      const int e = tid + i * 256;                 // n = a*128 + c = e < 4096
      const int a = e >> 7, c = e & 127;
      const float2 v = make_float2(Are[a * LDA_ + c] * sc,
                                   Aim[a * LDA_ + c] * sc);
      *(float2*)(op + (size_t)e * D_ + 2 * p) = v;
    }
  }
}

// ---------------------------------------------------------------------------
// Kernel 3: unpack packed spectrum, apply per-channel filter, repack for the
// packed inverse transform. In-place on the (k, 8192-k) pair.
// ---------------------------------------------------------------------------
__global__ __launch_bounds__(256) void spec_kernel(float2* __restrict__ Z,
                                                   const float* __restrict__ Fre,
                                                   const float* __restrict__ Fim)
{
  const int pi = blockIdx.y;
  const int k = blockIdx.x * 256 + threadIdx.x;
  if (k > 4096) return;
  const int h = (pi >> 5) & (H_ - 1);
  const int p = pi & (P_ - 1);
  const size_t zb = (size_t)pi * FFT_;
  const int km = (FFT_ - k) & (FFT_ - 1);
  const float2 Zk = Z[zb + k];
  const float2 Zm = Z[zb + km];
  // Xd  = 0.5*(Zk + conj(Zm));  Xd1 = -0.5i*(Zk - conj(Zm))
  const float xdr = 0.5f * (Zk.x + Zm.x), xdi = 0.5f * (Zk.y - Zm.y);
  const float x1r = 0.5f * (Zk.y + Zm.y), x1i = -0.5f * (Zk.x - Zm.x);
  const size_t fo = ((size_t)h * NF_ + k) * D_ + 2 * p;
  const float a0r = Fre[fo],     a0i = Fim[fo];
  const float a1r = Fre[fo + 1], a1i = Fim[fo + 1];
  const float Ydr = a0r * xdr - a0i * xdi, Ydi = a0r * xdi + a0i * xdr;
  const float Y1r = a1r * x1r - a1i * x1i, Y1i = a1r * x1i + a1i * x1r;
  // W[k] = Yd + i*Yd1 ; W[N-k] = conj(Yd) + i*conj(Yd1)
  Z[zb + k] = make_float2(Ydr - Y1i, Ydi + Y1r);
  if (km != k)
    Z[zb + km] = make_float2(Ydr + Y1i, Y1r - Ydi);
}

// ---------------------------------------------------------------------------
// Launch. Workspace layout (requires ~151 MB of d_ws, whole thing fits in the
// 192 MB L2): [Fre | Fim] filter planes (16 MB) then packed spectrum Z (128 MB).
// ---------------------------------------------------------------------------
extern "C" void kernel_launch(void* const* d_in, const int* in_sizes, int n_in,
                              void* d_out, int out_size, void* d_ws,
                              size_t ws_size, hipStream_t stream)
{
  const float* x  = (const float*)d_in[0];
  const float* w0 = (const float*)d_in[1];
  const float* b0 = (const float*)d_in[2];
  const float* w1 = (const float*)d_in[3];
  const float* b1 = (const float*)d_in[4];
  const float* w2 = (const float*)d_in[5];
  const float* b2 = (const float*)d_in[6];
  const float* w3 = (const float*)d_in[7];
  const float* b3 = (const float*)d_in[8];
  const float* wo = (const float*)d_in[9];
  const float* bo = (const float*)d_in[10];
  float* out = (float*)d_out;

  const size_t FSZ = (size_t)H_ * NF_ * D_;               // floats per plane
  float* Fre = (float*)d_ws;
  float* Fim = Fre + FSZ;
  float2* Z = (float2*)((char*)d_ws +
                        ((2 * FSZ * sizeof(float) + 255) & ~(size_t)255));

  rpe_kernel<<<257, 32, 0, stream>>>(w0, b0, w1, b1, w2, b2, w3, b3, wo, bo,
                                     Fre, Fim);
  const size_t smem = 4 * 64 * LDA_ * sizeof(float);      // 133,120 B / WG
  fft_kernel<0><<<B_ * H_ * P_, 256, smem, stream>>>(Z, x, nullptr);
  spec_kernel<<<dim3(17, B_ * H_ * P_), 256, 0, stream>>>(Z, Fre, Fim);
  fft_kernel<1><<<B_ * H_ * P_, 256, smem, stream>>>(Z, nullptr, out);
}